// GCNEncoder_7421703487979
// MI455X (gfx1250) — compile-verified
//
#include <hip/hip_runtime.h>

typedef __attribute__((ext_vector_type(2))) float v2f;
typedef __attribute__((ext_vector_type(8))) float v8f;

#define F_IN 128   // input feature width; hidden width is also 128
#define F_OUT 64
#define LDS_STRIDE 132  // padded transposed-W stride: 132 % 64 == 4 -> perfect bank cover

// ---------------------------------------------------------------------------
// Degree / normalization kernels
// ---------------------------------------------------------------------------
__global__ void init_deg_kernel(float* deg, int n) {
  int i = blockIdx.x * blockDim.x + threadIdx.x;
  if (i < n) deg[i] = 1.0f;  // self-loop contributes 1 to every node's degree
}

__global__ void count_deg_kernel(const long long* __restrict__ dst,
                                 float* __restrict__ deg, int ne) {
  int e = blockIdx.x * blockDim.x + threadIdx.x;
  if (e < ne) atomicAdd(&deg[(int)dst[e]], 1.0f);
}

__global__ void rsqrt_deg_kernel(float* deg, int n) {
  int i = blockIdx.x * blockDim.x + threadIdx.x;
  if (i < n) deg[i] = rsqrtf(deg[i]);  // deg >= 1 always (self-loop)
}

// ---------------------------------------------------------------------------
// Aggregation: out[i] = dis[i]^2 * feat[i]  (self-loop init), then
//              out[d] += dis[s]*dis[d] * feat[s] per edge (L2 scatter-add)
// ---------------------------------------------------------------------------
__global__ void init_selfloop_kernel(const float* __restrict__ feat,
                                     const float* __restrict__ dis,
                                     float* __restrict__ out, int n) {
  size_t idx = (size_t)blockIdx.x * blockDim.x + threadIdx.x;
  size_t total = (size_t)n * (F_IN / 4);
  if (idx >= total) return;
  int node = (int)(idx >> 5);  // 32 float4 quads per node
  float s = dis[node];
  s = s * s;
  float4 v = ((const float4*)feat)[idx];
  ((float4*)out)[idx] = make_float4(v.x * s, v.y * s, v.z * s, v.w * s);
}

__global__ void scatter_edges_kernel(const float* __restrict__ feat,
                                     const long long* __restrict__ src,
                                     const long long* __restrict__ dst,
                                     const float* __restrict__ dis,
                                     float* __restrict__ out, int ne) {
  // one wave (32 lanes) per edge; each lane handles 4 consecutive features
  int wave = (int)(((size_t)blockIdx.x * blockDim.x + threadIdx.x) >> 5);
  int lane = threadIdx.x & 31;
  if (wave >= ne) return;
  int s = (int)src[wave];
  int d = (int)dst[wave];
  float nrm = dis[s] * dis[d];
  float4 v = ((const float4*)(feat + (size_t)s * F_IN))[lane];
  float* o = out + (size_t)d * F_IN + lane * 4;
  atomicAdd(o + 0, v.x * nrm);
  atomicAdd(o + 1, v.y * nrm);
  atomicAdd(o + 2, v.z * nrm);
  atomicAdd(o + 3, v.w * nrm);
}

// ---------------------------------------------------------------------------
// WMMA fp32 GEMM:  O[nrows, NT*16] = act(A[nrows,128] @ W[128,NT*16] + bias)
//
// Block = 256 threads (8 waves). The block cooperatively stages W into LDS
// *transposed* (wt[col][k], stride LDS_STRIDE); then each wave computes one
// 16-row strip across ALL NT column tiles, so each A fragment feeds NT WMMAs:
//   per k-step: 1 global_load_b64 (A) + NT ds_load_b64 (B) + NT v_wmma.
//
// A (16x4 f32): lane&15 = M row; lane>>4 selects K offset {0,2}.
// B (4x16 f32): mirrored; wt layout makes the K pair a contiguous b64.
// C/D (16x16 f32): vgpr r, lane&15 = N; M = r + 8*(lane>>4).
// ---------------------------------------------------------------------------
template <bool RELU, int NT>
__global__ void wmma_gemm_kernel(const float* __restrict__ A,
                                 const float* __restrict__ W,
                                 const float* __restrict__ bias,
                                 float* __restrict__ O,
                                 int nrows) {
  constexpr int NCOLS = NT * 16;
  __shared__ float wt[NCOLS * LDS_STRIDE];

  // ---- stage W (128 x NCOLS, row-major) -> wt[col * LDS_STRIDE + k] ----
  for (int e = threadIdx.x; e < F_IN * NCOLS; e += 256) {
    int k = e >> (NT == 8 ? 7 : 6);  // e / NCOLS (NCOLS is 128 or 64)
    int col = e & (NCOLS - 1);
    wt[col * LDS_STRIDE + k] = W[e];
  }
  __syncthreads();

  int waveInBlk = threadIdx.x >> 5;
  int lane = threadIdx.x & 31;
  int tileM = blockIdx.x * 8 + waveInBlk;
  if (tileM * 16 >= nrows) return;  // wave-uniform (after the barrier)

  int kOff = (lane >> 4) * 2;  // 0 or 2
  int colL = lane & 15;
  const float* arow = A + (size_t)(tileM * 16 + colL) * F_IN + kOff;

  v8f c[NT] = {};
#pragma unroll 8
  for (int k = 0; k < F_IN; k += 4) {
    v2f a = *(const v2f*)(arow + k);
#pragma unroll
    for (int t = 0; t < NT; ++t) {
      v2f b = *(const v2f*)(&wt[(t * 16 + colL) * LDS_STRIDE + k + kOff]);
      c[t] = __builtin_amdgcn_wmma_f32_16x16x4_f32(
          /*neg_a=*/false, a, /*neg_b=*/false, b,
          /*c_mod=*/(short)0, c[t], /*reuse_a=*/false, /*reuse_b=*/false);
    }
  }

  // ---- bias + activation + store ----
  int row0 = tileM * 16 + (lane >> 4) * 8;
#pragma unroll
  for (int t = 0; t < NT; ++t) {
    int col = t * 16 + colL;
    float bv = bias[col];
#pragma unroll
    for (int r = 0; r < 8; ++r) {
      float v = c[t][r] + bv;
      if (RELU) v = fmaxf(v, 0.0f);
      O[(size_t)(row0 + r) * NCOLS + col] = v;
    }
  }
}

// ---------------------------------------------------------------------------
// Launch
// ---------------------------------------------------------------------------
extern "C" void kernel_launch(void* const* d_in, const int* in_sizes, int n_in,
                              void* d_out, int out_size, void* d_ws, size_t ws_size,
                              hipStream_t stream) {
  const float*     x    = (const float*)d_in[0];
  const long long* edge = (const long long*)d_in[1];
  const float*     W1   = (const float*)d_in[2];
  const float*     b1   = (const float*)d_in[3];
  const float*     Wmu  = (const float*)d_in[4];
  const float*     bmu  = (const float*)d_in[5];
  const float*     Wls  = (const float*)d_in[6];
  const float*     bls  = (const float*)d_in[7];

  int n  = in_sizes[0] / F_IN;  // 100000 nodes
  int ne = in_sizes[1] / 2;     // 1,000,000 edges
  const long long* src = edge;       // edge_index[0]
  const long long* dst = edge + ne;  // edge_index[1]

  // workspace layout: dis [n] | agg [n*128] | h [n*128]
  char* ws = (char*)d_ws;
  float* dis = (float*)ws;
  size_t off = ((size_t)n * 4 + 1023) & ~(size_t)1023;
  float* agg = (float*)(ws + off);
  off += (size_t)n * F_IN * 4;
  float* h = (float*)(ws + off);

  float* mu = (float*)d_out;
  float* ls = mu + (size_t)n * F_OUT;

  const int T = 256;
  // 1. degrees + symmetric normalization
  init_deg_kernel<<<(n + T - 1) / T, T, 0, stream>>>(dis, n);
  count_deg_kernel<<<(ne + T - 1) / T, T, 0, stream>>>(dst, dis, ne);
  rsqrt_deg_kernel<<<(n + T - 1) / T, T, 0, stream>>>(dis, n);

  // 2. aggX = Agg(x)
  size_t quads = (size_t)n * (F_IN / 4);
  init_selfloop_kernel<<<(int)((quads + T - 1) / T), T, 0, stream>>>(x, dis, agg, n);
  scatter_edges_kernel<<<(int)(((size_t)ne * 32 + T - 1) / T), T, 0, stream>>>(
      x, src, dst, dis, agg, ne);

  // 3. h = relu(aggX @ W1 + b1)   [100000 x 128 @ 128 x 128]
  {
    int tilesM = n / 16;                 // 6250
    int blocks = (tilesM + 7) / 8;       // 8 waves (strips) per block
    wmma_gemm_kernel<true, 8><<<blocks, T, 0, stream>>>(agg, W1, b1, h, n);
  }

  // 4. aggH = Agg(h)  (reuse agg buffer)
  init_selfloop_kernel<<<(int)((quads + T - 1) / T), T, 0, stream>>>(h, dis, agg, n);
  scatter_edges_kernel<<<(int)(((size_t)ne * 32 + T - 1) / T), T, 0, stream>>>(
      h, src, dst, dis, agg, ne);

  // 5. mu / logstd GEMMs   [100000 x 128 @ 128 x 64]
  {
    int tilesM = n / 16;
    int blocks = (tilesM + 7) / 8;
    wmma_gemm_kernel<false, 4><<<blocks, T, 0, stream>>>(agg, Wmu, bmu, mu, n);
    wmma_gemm_kernel<false, 4><<<blocks, T, 0, stream>>>(agg, Wls, bls, ls, n);
  }
}